// RepPointsGenerator_24343874633950
// MI455X (gfx1250) — compile-verified
//
#include <hip/hip_runtime.h>
#include <hip/hip_bf16.h>
#include <cstdint>

// ---------------------------------------------------------------------------
// RepPoints proposal generator for MI455X (gfx1250, wave32).
//  - k_match: GT<->center squared-distance matrix via V_WMMA_F32_16X16X4_F32
//    (per-level compile-time-constant sweeps, ballot-skipped, branchless argmin)
//  - exact top-k by 32-step binary search on order-preserving uint keys
//  - bitmask NMS: parallel mask build (async LDS staging) + one-wave scan
// ---------------------------------------------------------------------------

#define N_IMG   16
#define X_TOT   48960
#define N_GT    128
#define K_NMS   3576      // 1000+1000+1000+576
#define KW      112       // ceil(3576/32)
#define IMGSZ   1536.0f
#define NMS_TH  0.7f

typedef __attribute__((ext_vector_type(2))) float v2f;
typedef __attribute__((ext_vector_type(8))) float v8f;
typedef __attribute__((__vector_size__(4 * sizeof(int)))) int v4i_t;
typedef __attribute__((address_space(3))) v4i_t lds_v4i_t;

#if defined(__has_builtin)
#if __has_builtin(__builtin_amdgcn_global_load_async_to_lds_b128) && \
    __has_builtin(__builtin_amdgcn_s_wait_asynccnt)
#define HAVE_ASYNC_LDS 1
#endif
#endif

__device__ inline int   lvl_of(int x)    { return (x < 36864) ? 0 : (x < 46080) ? 1 : (x < 48384) ? 2 : 3; }
__device__ inline int   lvl_base(int li) { return (li == 0) ? 0 : (li == 1) ? 36864 : (li == 2) ? 46080 : 48384; }
__device__ inline int   lvl_w(int li)    { return 192 >> li; }
__device__ inline float lvl_s(int li)    { return (float)(8 << li); }

__device__ inline unsigned f2key(float f) {
  unsigned b = __float_as_uint(f);
  return (b & 0x80000000u) ? ~b : (b | 0x80000000u);
}
__device__ inline float key2f(unsigned u) {
  return (u & 0x80000000u) ? __uint_as_float(u & 0x7FFFFFFFu) : __uint_as_float(~u);
}

// ---------------------------------------------------------------------------
// Level sweep for k_match: one WMMA f32 16x16x4 per 16-center tile.
//   A row m = (-2gx, -2gy, |g|^2, 1),  B col n = (cx, cy, 1, |c|^2)
// Wave32 layouts (ISA 7.12.2):
//   A/B: lane l holds row/col l%16, VGPR0/1 = K {0,1} (l<16) or {2,3} (l>=16)
//   C:   VGPR r, lane l -> M = r + 8*(l>=16), N = l%16
// EXEC stays all-ones: the level-skip branch is wave-uniform (ballot result),
// and the argmin update is pure v_cndmask selects.
// ---------------------------------------------------------------------------
template <int LI>
__device__ inline void scan_level(unsigned needmask, v2f a, int half, int m,
                                  const int* __restrict__ lo8,
                                  const int* __restrict__ hi8,
                                  float* __restrict__ best,
                                  int* __restrict__ bidx) {
  if (!(needmask & (1u << LI))) return;           // wave-uniform skip
  constexpr int   W    = 192 >> LI;
  constexpr float S    = (float)(8 << LI);
  constexpr int   BASE = (LI == 0) ? 0 : (LI == 1) ? 36864 : (LI == 2) ? 46080 : 48384;
  const int t0 = BASE / 16, t1 = (BASE + W * W) / 16;  // level edges are 16-aligned
  for (int t = t0; t < t1; ++t) {
    const int nidx = t * 16 + m;
    const int ci = nidx - BASE;
    const float cx = ((float)(ci % W) + 0.5f) * S;
    const float cy = ((float)(ci / W) + 0.5f) * S;
    v2f b;
    if (half == 0) { b.x = cx;   b.y = cy; }
    else           { b.x = 1.0f; b.y = cx * cx + cy * cy; }
    v8f c = {0.f, 0.f, 0.f, 0.f, 0.f, 0.f, 0.f, 0.f};
    v8f d = __builtin_amdgcn_wmma_f32_16x16x4_f32(false, a, false, b,
                                                  (short)0, c, false, false);
#pragma unroll
    for (int r = 0; r < 8; ++r) {
      const float dd = d[r];
      const bool ok = (nidx >= lo8[r]) && (nidx < hi8[r]) && (dd < best[r]);
      best[r] = ok ? dd : best[r];
      bidx[r] = ok ? nidx : bidx[r];
    }
  }
}

__global__ __launch_bounds__(256) void k_match(const float* __restrict__ gt,
                                               int* __restrict__ argmin_out) {
  const int n    = blockIdx.x;
  const int lane = threadIdx.x & 31;
  const int wave = threadIdx.x >> 5;     // 0..7 -> GT group of 16
  const int half = lane >> 4;            // 0 = K{0,1}, 1 = K{2,3}
  const int m    = lane & 15;

  // Load this lane's GT (row m of the A tile).
  const int g = wave * 16 + m;
  const float* gb = gt + ((size_t)n * N_GT + g) * 4;
  float x1 = gb[0], y1 = gb[1], x2 = gb[2], y2 = gb[3];
  float gx = 0.5f * (x1 + x2), gy = 0.5f * (y1 + y2);
  float gw = fmaxf(x2 - x1, 1e-6f), gh = fmaxf(y2 - y1, 1e-6f);
  float lvl = floorf(0.5f * log2f(gw * gh) - 2.0f);   // log2(SCALE)=2
  lvl = fminf(fmaxf(lvl, 3.0f), 6.0f);
  int li = (int)lvl - 3;
  int rlo = lvl_base(li);
  int rhi = rlo + lvl_w(li) * lvl_w(li);

  // Which levels does this wave's GT group touch? (uniform per wave)
  unsigned needmask = 0;
#pragma unroll
  for (int L = 0; L < 4; ++L) {
    unsigned long long bl = __ballot(li == L);
    if (bl) needmask |= (1u << L);
  }

  // Per-C-row level ranges, gathered by shuffle (no LDS, no barrier).
  int lo8[8], hi8[8];
#pragma unroll
  for (int r = 0; r < 8; ++r) {
    int M = r + half * 8;
    lo8[r] = __shfl(rlo, M, 32);
    hi8[r] = __shfl(rhi, M, 32);
  }

  v2f a;
  if (half == 0) { a.x = -2.0f * gx;        a.y = -2.0f * gy; }
  else           { a.x = gx * gx + gy * gy; a.y = 1.0f; }

  float best[8];
  int   bidx[8];
#pragma unroll
  for (int r = 0; r < 8; ++r) { best[r] = __builtin_inff(); bidx[r] = 0; }

  scan_level<0>(needmask, a, half, m, lo8, hi8, best, bidx);
  scan_level<1>(needmask, a, half, m, lo8, hi8, best, bidx);
  scan_level<2>(needmask, a, half, m, lo8, hi8, best, bidx);
  scan_level<3>(needmask, a, half, m, lo8, hi8, best, bidx);

  // Reduce across the 16 lanes sharing each row (xor stays within halves).
#pragma unroll
  for (int off = 1; off < 16; off <<= 1) {
#pragma unroll
    for (int r = 0; r < 8; ++r) {
      float od = __shfl_xor(best[r], off, 32);
      int   oi = __shfl_xor(bidx[r], off, 32);
      bool tk = (od < best[r]) || (od == best[r] && oi < bidx[r]);
      best[r] = tk ? od : best[r];
      bidx[r] = tk ? oi : bidx[r];
    }
  }
  if ((lane & 15) == 0) {
#pragma unroll
    for (int r = 0; r < 8; ++r)
      argmin_out[n * N_GT + wave * 16 + r + half * 8] = bidx[r];
  }
}

// ---------------------------------------------------------------------------
__global__ __launch_bounds__(256) void k_assign_init(int* __restrict__ assign) {
  int idx = blockIdx.x * 256 + threadIdx.x;
  if (idx < N_IMG * X_TOT) assign[idx] = -1;
}

// Sequential last-writer-wins scatter, matching .at[idx].set order.
__global__ void k_assign_scatter(const int* __restrict__ argmin, int* __restrict__ assign) {
  int n = threadIdx.x;
  if (n >= N_IMG) return;
  for (int g = 0; g < N_GT; ++g)
    assign[n * X_TOT + argmin[n * N_GT + g]] = g;
}

// ---------------------------------------------------------------------------
// Decode boxes + focal loss + loc-loss partial sums.
// scal layout: [0]=pos_cnt(int) [1]=pos_sum [2]=loc_sum [3]=neg_sum [4]=neg_cnt(int)
// ---------------------------------------------------------------------------
__global__ __launch_bounds__(256) void k_decode_focal(
    const float* __restrict__ lg0, const float* __restrict__ lg1,
    const float* __restrict__ lg2, const float* __restrict__ lg3,
    const float* __restrict__ dl0, const float* __restrict__ dl1,
    const float* __restrict__ dl2, const float* __restrict__ dl3,
    const float* __restrict__ gt, const int* __restrict__ assign,
    float* __restrict__ pred, float* __restrict__ negval, float* __restrict__ scal) {
  __shared__ float s_ps, s_loc; __shared__ int s_pc;
  if (threadIdx.x == 0) { s_ps = 0.f; s_loc = 0.f; s_pc = 0; }
  __syncthreads();

  int idx = blockIdx.x * 256 + threadIdx.x;
  float my_ps = 0.f, my_loc = 0.f; int my_pc = 0;
  if (idx < N_IMG * X_TOT) {
    int n = idx / X_TOT, x = idx % X_TOT;
    int li = lvl_of(x), base = lvl_base(li), w = lvl_w(li);
    float s = lvl_s(li);
    int i = x - base, hw = w * w;
    float cx = ((float)(i % w) + 0.5f) * s;
    float cy = ((float)(i / w) + 0.5f) * s;
    const float* lg = (li == 0) ? lg0 : (li == 1) ? lg1 : (li == 2) ? lg2 : lg3;
    const float* dl = (li == 0) ? dl0 : (li == 1) ? dl1 : (li == 2) ? dl2 : dl3;
    float z  = lg[n * hw + i];
    float d0 = dl[((size_t)(n * 4 + 0)) * hw + i];
    float d1 = dl[((size_t)(n * 4 + 1)) * hw + i];
    float d2 = dl[((size_t)(n * 4 + 2)) * hw + i];
    float d3 = dl[((size_t)(n * 4 + 3)) * hw + i];
    float b0 = cx + d0 * s, b1 = cy + d1 * s, b2 = cx + d2 * s, b3 = cy + d3 * s;
    size_t po = (size_t)idx * 4;
    pred[po + 0] = b0; pred[po + 1] = b1; pred[po + 2] = b2; pred[po + 3] = b3;

    int g = assign[idx];
    float t = (g >= 0) ? 1.0f : 0.0f;
    float p = 1.0f / (1.0f + expf(-z));
    float ce = fmaxf(z, 0.0f) - z * t + log1pf(expf(-fabsf(z)));
    float p_t = p * t + (1.0f - p) * (1.0f - t);
    float a_t = 0.25f * t + 0.75f * (1.0f - t);
    float om = 1.0f - p_t;
    float cls = a_t * ce * om * om;

    if (g >= 0) {
      my_pc = 1; my_ps = cls; negval[idx] = 0.0f;
      float inv = 1.0f / (s * 4.0f);
      const float* gbp = gt + ((size_t)n * N_GT + g) * 4;
      float pb4[4] = {b0, b1, b2, b3};
      float acc = 0.f;
#pragma unroll
      for (int c = 0; c < 4; ++c) {
        float d = fabsf(pb4[c] * inv - gbp[c] * inv);
        acc += (d < 0.11f) ? (0.5f * d * d / 0.11f) : (d - 0.055f);
      }
      my_loc = acc;
    } else {
      negval[idx] = cls;
    }
  }
  atomicAdd(&s_ps, my_ps); atomicAdd(&s_loc, my_loc); atomicAdd(&s_pc, my_pc);
  __syncthreads();
  if (threadIdx.x == 0) {
    atomicAdd(&scal[1], s_ps);
    atomicAdd(&scal[2], s_loc);
    atomicAdd((int*)scal, s_pc);
  }
}

// ---------------------------------------------------------------------------
// OHEM negative mining: exact top-k sum via binary search on uint-ordered keys.
// ---------------------------------------------------------------------------
__global__ __launch_bounds__(1024) void k_negsel(const float* __restrict__ negval,
                                                 float* __restrict__ scal) {
  __shared__ unsigned s_cnt; __shared__ float s_sum; __shared__ int s_c2;
  const int NX = N_IMG * X_TOT;
  int pos_raw = ((const int*)scal)[0];
  long pos_c = (pos_raw < 1) ? 1 : pos_raw;
  long neg_total = (long)NX - (long)pos_raw;
  long kk = (neg_total < 3 * pos_c) ? neg_total : 3 * pos_c;
  if (kk < 1) kk = 1;
  const unsigned k = (unsigned)kk;

  unsigned lo = 0u, hi = 0xFFFFFFFFu;
  for (int it = 0; it < 32; ++it) {
    unsigned mid = lo + ((hi - lo) >> 1);
    if (threadIdx.x == 0) s_cnt = 0;
    __syncthreads();
    unsigned c = 0;
    for (int i = threadIdx.x; i < NX; i += 1024)
      if (f2key(negval[i]) > mid) c++;
    atomicAdd(&s_cnt, c);
    __syncthreads();
    unsigned tot = s_cnt;
    __syncthreads();
    if (tot < k) hi = mid; else lo = mid + 1;
  }
  const unsigned T = hi;   // k-th largest key

  if (threadIdx.x == 0) { s_sum = 0.f; s_c2 = 0; }
  __syncthreads();
  float mysum = 0.f; int myc = 0;
  for (int i = threadIdx.x; i < NX; i += 1024) {
    float v = negval[i];
    if (f2key(v) > T) { mysum += v; myc++; }
  }
  atomicAdd(&s_sum, mysum); atomicAdd(&s_c2, myc);
  __syncthreads();
  if (threadIdx.x == 0) {
    float fT = key2f(T);
    scal[3] = s_sum + (float)((long)k - (long)s_c2) * fT;
    ((int*)scal)[4] = (int)k;
  }
}

// ---------------------------------------------------------------------------
// Per-level top-k selection (set only; order irrelevant pre-global-sort).
// ---------------------------------------------------------------------------
__global__ __launch_bounds__(256) void k_topk(
    const float* __restrict__ lg0, const float* __restrict__ lg1,
    const float* __restrict__ lg2, const float* __restrict__ lg3,
    float* __restrict__ tsel_s, int* __restrict__ tsel_x) {
  const int n = blockIdx.x >> 2, li = blockIdx.x & 3;
  const int w = lvl_w(li), hw = w * w;
  const float* lgp = (li == 0) ? lg0 : (li == 1) ? lg1 : (li == 2) ? lg2 : lg3;
  const float* lg = lgp + (size_t)n * hw;
  const int k = (li < 3) ? 1000 : 576;
  const int sbase = li * 1000;
  const int xbase = lvl_base(li);

  if (k == hw) {  // level 3: take everything in index order
    for (int i = threadIdx.x; i < hw; i += 256) {
      tsel_s[n * K_NMS + sbase + i] = lg[i];
      tsel_x[n * K_NMS + sbase + i] = xbase + i;
    }
    return;
  }

  __shared__ unsigned s_cnt; __shared__ int s_c1;
  unsigned lo = 0u, hi = 0xFFFFFFFFu;
  for (int it = 0; it < 32; ++it) {
    unsigned mid = lo + ((hi - lo) >> 1);
    if (threadIdx.x == 0) s_cnt = 0;
    __syncthreads();
    unsigned c = 0;
    for (int i = threadIdx.x; i < hw; i += 256)
      if (f2key(lg[i]) > mid) c++;
    atomicAdd(&s_cnt, c);
    __syncthreads();
    unsigned tot = s_cnt;
    __syncthreads();
    if (tot < (unsigned)k) hi = mid; else lo = mid + 1;
  }
  const unsigned T = hi;

  if (threadIdx.x == 0) s_c1 = 0;
  __syncthreads();
  for (int i = threadIdx.x; i < hw; i += 256) {
    if (f2key(lg[i]) > T) {
      int p = atomicAdd(&s_c1, 1);
      tsel_s[n * K_NMS + sbase + p] = lg[i];
      tsel_x[n * K_NMS + sbase + p] = xbase + i;
    }
  }
  __syncthreads();
  if (threadIdx.x == 0) {  // fill ties (== T) by lowest index, like lax.top_k
    int c = s_c1, rem = k - c;
    for (int i = 0; i < hw && rem > 0; ++i) {
      if (f2key(lg[i]) == T) {
        tsel_s[n * K_NMS + sbase + c] = lg[i];
        tsel_x[n * K_NMS + sbase + c] = xbase + i;
        c++; rem--;
      }
    }
  }
}

// ---------------------------------------------------------------------------
__global__ __launch_bounds__(256) void k_prep(const float* __restrict__ tsel_s,
                                              const int* __restrict__ tsel_x,
                                              const float* __restrict__ pred,
                                              float* __restrict__ nms_s,
                                              float* __restrict__ nms_b) {
  int idx = blockIdx.x * 256 + threadIdx.x;
  if (idx >= N_IMG * K_NMS) return;
  int n = idx / K_NMS;
  int x = tsel_x[idx];
  float b[4];
#pragma unroll
  for (int c = 0; c < 4; ++c)
    b[c] = fminf(fmaxf(pred[((size_t)n * X_TOT + x) * 4 + c], 0.0f), IMGSZ);
  bool valid = (b[2] - b[0] >= 0.0f) && (b[3] - b[1] >= 0.0f);
  nms_s[idx] = valid ? tsel_s[idx] : -__builtin_inff();
#pragma unroll
  for (int c = 0; c < 4; ++c) nms_b[(size_t)idx * 4 + c] = b[c];
}

// ---------------------------------------------------------------------------
// Per-image bitonic sort of (score desc, slot asc) over padded-4096 entries.
// ---------------------------------------------------------------------------
__global__ __launch_bounds__(512) void k_sort(const float* __restrict__ nms_s,
                                              float* __restrict__ sscore,
                                              int* __restrict__ spos) {
  __shared__ float key[4096];
  __shared__ int   pos[4096];
  const int n = blockIdx.x;
  for (int i = threadIdx.x; i < 4096; i += 512) {
    if (i < K_NMS) { key[i] = nms_s[n * K_NMS + i]; pos[i] = i; }
    else           { key[i] = -__builtin_inff();    pos[i] = 100000 + i; }
  }
  __syncthreads();
  for (int ksz = 2; ksz <= 4096; ksz <<= 1) {
    for (int j = ksz >> 1; j > 0; j >>= 1) {
      for (int i = threadIdx.x; i < 4096; i += 512) {
        int l = i ^ j;
        if (l > i) {
          float ka = key[i], kb = key[l]; int pa = pos[i], pb = pos[l];
          bool before = (ka > kb) || (ka == kb && pa < pb);
          bool up = ((i & ksz) == 0);
          bool sw = up ? !before : before;
          if (sw) { key[i] = kb; key[l] = ka; pos[i] = pb; pos[l] = pa; }
        }
      }
      __syncthreads();
    }
  }
  for (int i = threadIdx.x; i < K_NMS; i += 512) {
    sscore[n * K_NMS + i] = key[i];
    spos[n * K_NMS + i]   = pos[i];
  }
}

// ---------------------------------------------------------------------------
__global__ __launch_bounds__(256) void k_gather(const int* __restrict__ spos,
                                                const float* __restrict__ nms_b,
                                                float* __restrict__ bo,
                                                float* __restrict__ boff) {
  int idx = blockIdx.x * 256 + threadIdx.x;
  if (idx >= N_IMG * K_NMS) return;
  int n = idx / K_NMS;
  int p = spos[idx];
  int lvl = (p < 1000) ? 0 : (p < 2000) ? 1 : (p < 3000) ? 2 : 3;
  float offv = (float)lvl * (IMGSZ + 2.0f);
#pragma unroll
  for (int c = 0; c < 4; ++c) {
    float v = nms_b[((size_t)n * K_NMS + p) * 4 + c];
    bo[(size_t)idx * 4 + c]   = v;
    boff[(size_t)idx * 4 + c] = v + offv;
  }
}

// ---------------------------------------------------------------------------
// NMS suppression mask matrix: bit j of mask[n][i][j/32] = (IoU(i,j) > 0.7).
// Boxes staged into LDS (57KB of the 320KB WGP LDS) via gfx1250 async-to-LDS
// when the toolchain exposes it, else plain loads (+ global_prefetch_b8).
// ---------------------------------------------------------------------------
__global__ __launch_bounds__(256) void k_mask(const float* __restrict__ boff,
                                              unsigned* __restrict__ mask) {
  __shared__ float sb[K_NMS * 4];
  const int n = blockIdx.y;
  const float* src = boff + (size_t)n * K_NMS * 4;
#if defined(HAVE_ASYNC_LDS)
  {
    for (int i = threadIdx.x; i < K_NMS; i += 256) {
      __builtin_amdgcn_global_load_async_to_lds_b128(
          (v4i_t*)(src + (size_t)i * 4),
          (lds_v4i_t*)(sb + (size_t)i * 4),
          0, 0);
    }
    __builtin_amdgcn_s_wait_asynccnt(0);
  }
#else
  for (int i = threadIdx.x; i < K_NMS * 4; i += 256) {
    __builtin_prefetch(&src[i], 0, 0);  // global_prefetch_b8
    sb[i] = src[i];
  }
#endif
  __syncthreads();
  const int i = blockIdx.x * 256 + threadIdx.x;
  if (i >= K_NMS) return;
  const float ax1 = sb[i * 4 + 0], ay1 = sb[i * 4 + 1];
  const float ax2 = sb[i * 4 + 2], ay2 = sb[i * 4 + 3];
  const float aarea = fmaxf(ax2 - ax1, 0.f) * fmaxf(ay2 - ay1, 0.f);
  unsigned* mrow = mask + ((size_t)n * K_NMS + i) * KW;
  for (int wd = 0; wd < KW; ++wd) {
    unsigned bits = 0;
#pragma unroll 4
    for (int b = 0; b < 32; ++b) {
      int j = wd * 32 + b;
      if (j < K_NMS) {
        float bx1 = sb[j * 4 + 0], by1 = sb[j * 4 + 1];
        float bx2 = sb[j * 4 + 2], by2 = sb[j * 4 + 3];
        float barea = fmaxf(bx2 - bx1, 0.f) * fmaxf(by2 - by1, 0.f);
        float ix = fminf(ax2, bx2) - fmaxf(ax1, bx1);
        float iy = fminf(ay2, by2) - fmaxf(ay1, by1);
        float inter = fmaxf(ix, 0.f) * fmaxf(iy, 0.f);
        float iou = inter / fmaxf(aarea + barea - inter, 1e-6f);
        if (iou > NMS_TH) bits |= (1u << b);
      }
    }
    mrow[wd] = bits;
  }
}

// ---------------------------------------------------------------------------
// One wave per image: sequential greedy scan over the sorted list using the
// precomputed suppression bitmasks, with fused emission of the final top-1000.
// rem[] = 112 removal words distributed across the 32 lanes (4 words/lane).
// ---------------------------------------------------------------------------
__global__ __launch_bounds__(32) void k_scan(const float* __restrict__ sscore,
                                             const float* __restrict__ bo,
                                             const unsigned* __restrict__ mask,
                                             float* __restrict__ out_boxes,
                                             float* __restrict__ out_scores) {
  const int n = blockIdx.x;
  const int lane = threadIdx.x;
  unsigned rem[4] = {0u, 0u, 0u, 0u};
  __shared__ unsigned char keepf[K_NMS];
  int outcnt = 0;
  const float NEGINF = -__builtin_inff();

  for (int i = 0; i < K_NMS; ++i) {
    const int w = i >> 5, b = i & 31;
    const int src = w & 31, q = w >> 5;
    unsigned v = (unsigned)__shfl((int)rem[q], src, 32);
    const bool sup = (v >> b) & 1u;
    const float sc = sscore[n * K_NMS + i];
    if (!sup) {
      keepf[i] = 1;
      const unsigned* mrow = mask + ((size_t)n * K_NMS + i) * KW;
#pragma unroll
      for (int q2 = 0; q2 < 4; ++q2) {
        int w2 = lane + q2 * 32;
        if (w2 < KW) rem[q2] |= mrow[w2];
      }
      if (sc > NEGINF && outcnt < 1000) {
        if (lane == 0) {
          out_scores[n * 1000 + outcnt] = sc;
#pragma unroll
          for (int c = 0; c < 4; ++c)
            out_boxes[((size_t)n * 1000 + outcnt) * 4 + c] =
                bo[((size_t)n * K_NMS + i) * 4 + c];
        }
        outcnt++;
      }
    } else {
      keepf[i] = 0;
    }
  }
  __syncthreads();
  // Tail: entries with -inf final score (suppressed or invalid), index order.
  for (int i = 0; i < K_NMS && outcnt < 1000; ++i) {
    const float sc = sscore[n * K_NMS + i];
    const bool emitted = keepf[i] && (sc > NEGINF);
    if (!emitted) {
      if (lane == 0) {
        out_scores[n * 1000 + outcnt] = NEGINF;
#pragma unroll
        for (int c = 0; c < 4; ++c)
          out_boxes[((size_t)n * 1000 + outcnt) * 4 + c] =
              bo[((size_t)n * K_NMS + i) * 4 + c];
      }
      outcnt++;
    }
  }
}

// ---------------------------------------------------------------------------
__global__ void k_final(const float* __restrict__ scal, float* __restrict__ out) {
  int pos_raw = ((const int*)scal)[0];
  float pc = (float)((pos_raw < 1) ? 1 : pos_raw);
  float pos_loss = scal[1] / pc;
  int nk = ((const int*)scal)[4]; if (nk < 1) nk = 1;
  float neg_loss = scal[3] / (float)nk;
  out[80000] = pos_loss + neg_loss;
  out[80001] = scal[2] / (4.0f * pc);
}

// ---------------------------------------------------------------------------
extern "C" void kernel_launch(void* const* d_in, const int* in_sizes, int n_in,
                              void* d_out, int out_size, void* d_ws, size_t ws_size,
                              hipStream_t stream) {
  (void)in_sizes; (void)n_in; (void)out_size; (void)ws_size;
  const float* lg[4] = {(const float*)d_in[0], (const float*)d_in[1],
                        (const float*)d_in[2], (const float*)d_in[3]};
  const float* dl[4] = {(const float*)d_in[4], (const float*)d_in[5],
                        (const float*)d_in[6], (const float*)d_in[7]};
  const float* gt = (const float*)d_in[8];
  float* out = (float*)d_out;

  // ---- workspace bump allocator (256B aligned) ----
  char* base = (char*)d_ws;
  size_t off = 0;
  auto alloc = [&](size_t bytes) -> void* {
    void* p = (void*)(base + off);
    off += (bytes + 255) & ~(size_t)255;
    return p;
  };
  float*    pred    = (float*)   alloc((size_t)N_IMG * X_TOT * 4 * sizeof(float));
  int*      assign  = (int*)     alloc((size_t)N_IMG * X_TOT * sizeof(int));
  float*    negval  = (float*)   alloc((size_t)N_IMG * X_TOT * sizeof(float));
  int*      argmin  = (int*)     alloc((size_t)N_IMG * N_GT * sizeof(int));
  float*    tsel_s  = (float*)   alloc((size_t)N_IMG * K_NMS * sizeof(float));
  int*      tsel_x  = (int*)     alloc((size_t)N_IMG * K_NMS * sizeof(int));
  float*    nms_s   = (float*)   alloc((size_t)N_IMG * K_NMS * sizeof(float));
  float*    nms_b   = (float*)   alloc((size_t)N_IMG * K_NMS * 4 * sizeof(float));
  float*    sscore  = (float*)   alloc((size_t)N_IMG * K_NMS * sizeof(float));
  int*      spos    = (int*)     alloc((size_t)N_IMG * K_NMS * sizeof(int));
  float*    bo      = (float*)   alloc((size_t)N_IMG * K_NMS * 4 * sizeof(float));
  float*    boff    = (float*)   alloc((size_t)N_IMG * K_NMS * 4 * sizeof(float));
  unsigned* mask    = (unsigned*)alloc((size_t)N_IMG * K_NMS * KW * sizeof(unsigned));
  float*    scal    = (float*)   alloc(64);

  (void)hipMemsetAsync(scal, 0, 64, stream);

  const int NX_BLKS = (N_IMG * X_TOT + 255) / 256;   // 3060
  const int NK_BLKS = (N_IMG * K_NMS + 255) / 256;   // 224

  k_match<<<N_IMG, 256, 0, stream>>>(gt, argmin);
  k_assign_init<<<NX_BLKS, 256, 0, stream>>>(assign);
  k_assign_scatter<<<1, N_IMG, 0, stream>>>(argmin, assign);
  k_decode_focal<<<NX_BLKS, 256, 0, stream>>>(lg[0], lg[1], lg[2], lg[3],
                                              dl[0], dl[1], dl[2], dl[3],
                                              gt, assign, pred, negval, scal);
  k_negsel<<<1, 1024, 0, stream>>>(negval, scal);
  k_topk<<<N_IMG * 4, 256, 0, stream>>>(lg[0], lg[1], lg[2], lg[3], tsel_s, tsel_x);
  k_prep<<<NK_BLKS, 256, 0, stream>>>(tsel_s, tsel_x, pred, nms_s, nms_b);
  k_sort<<<N_IMG, 512, 0, stream>>>(nms_s, sscore, spos);
  k_gather<<<NK_BLKS, 256, 0, stream>>>(spos, nms_b, bo, boff);
  k_mask<<<dim3((K_NMS + 255) / 256, N_IMG), 256, 0, stream>>>(boff, mask);
  k_scan<<<N_IMG, 32, 0, stream>>>(sscore, bo, mask, out, out + 64000);
  k_final<<<1, 1, 0, stream>>>(scal, out);
}